// Attn_22814866276758
// MI455X (gfx1250) — compile-verified
//
#include <hip/hip_runtime.h>
#include <hip/hip_bf16.h>

#define SEQ_LEN 32768
#define HID 1024

#define KC 64                    // K-chunk per pipeline stage
#define NCHUNK (HID / KC)        // 16
#define ROWS_PER_WG 64           // 4 waves * 16 rows
#define BLOCK 128
#define LDS_STRIDE (KC + 4)      // 68 floats: (m*68)%64 = 4m -> conflict-free A reads
#define F4_PER_ROW (KC / 4)      // 16 float4 per staged row
#define LD_PER_THREAD 8          // 64 rows * 16 f4 / 128 threads

#define USE_ASYNC_LDS 1          // gfx1250 GLOBAL_LOAD_ASYNC_TO_LDS_B128 path

typedef __attribute__((ext_vector_type(2))) float v2f;
typedef __attribute__((ext_vector_type(8))) float v8f;

// ---------------------------------------------------------------------------
// Kernel A: v = W^T h   (v[j] = sum_d W[d*H + j] * h[d]); 4 MB read, trivial.
// ---------------------------------------------------------------------------
__global__ void __launch_bounds__(256)
proj_vec_kernel(const float* __restrict__ W,
                const float* __restrict__ h,
                float* __restrict__ v) {
  const int j = blockIdx.x * 256 + threadIdx.x;
  float acc = 0.f;
#pragma unroll 4
  for (int d = 0; d < HID; ++d)
    acc += W[(size_t)d * HID + j] * h[d];
  v[j] = acc;
}

// ---------------------------------------------------------------------------
// Kernel B: energies[s] = enc[s,:] . v  via V_WMMA_F32_16X16X4_F32.
//   Staging: double-buffered GLOBAL_LOAD_ASYNC_TO_LDS_B128 (ASYNCcnt),
//   8 async instrs per wave per chunk -> wait asynccnt<=8 isolates chunk c.
// ---------------------------------------------------------------------------
__global__ void __launch_bounds__(BLOCK)
energies_kernel(const float* __restrict__ enc,
                const float* __restrict__ v,
                float* __restrict__ energies) {
  __shared__ alignas(16) float tile[2][ROWS_PER_WG * LDS_STRIDE];
  __shared__ alignas(16) float vsh[HID];

  const int tid     = threadIdx.x;
  const int lane    = tid & 31;
  const int wave    = tid >> 5;                 // 0..3
  const int rowBase = blockIdx.x * ROWS_PER_WG;

  const int koff = (lane < 16) ? 0 : 2;         // A/B lane -> K sub-offset
  const int m    = lane & 15;                   // A lane -> row within 16

  // Preload entire v (4 KB) into LDS once.
  ((float4*)vsh)[tid]         = ((const float4*)v)[tid];
  ((float4*)vsh)[tid + BLOCK] = ((const float4*)v)[tid + BLOCK];

  // Per-thread staging slots: LD_PER_THREAD b128 transfers per chunk.
  // idx = i*BLOCK + tid; row = idx/16, c4 = idx%16  (wave-contiguous rows).
  v8f acc = {};

  auto issue_chunk = [&](int k0, int buf) {
#pragma unroll
    for (int i = 0; i < LD_PER_THREAD; ++i) {
      const int idx = i * BLOCK + tid;
      const int r   = idx >> 4;
      const int c4  = idx & 15;
      const float* g = enc + (size_t)(rowBase + r) * HID + k0 + (c4 << 2);
      float* l = &tile[buf][r * LDS_STRIDE + (c4 << 2)];
#if USE_ASYNC_LDS
      const unsigned laddr = (unsigned)(size_t)l;   // low 32 bits = LDS offset
      asm volatile("global_load_async_to_lds_b128 %0, %1, off"
                   :: "v"(laddr), "v"(g) : "memory");
#else
      // sync fallback: batched loads then stores (done below instead)
      (void)g; (void)l;
#endif
    }
#if !USE_ASYNC_LDS
    float4 tmp[LD_PER_THREAD];
#pragma unroll
    for (int i = 0; i < LD_PER_THREAD; ++i) {
      const int idx = i * BLOCK + tid;
      tmp[i] = ((const float4*)enc)[(size_t)(rowBase + (idx >> 4)) * (HID / 4)
                                    + (k0 >> 2) + (idx & 15)];
    }
#pragma unroll
    for (int i = 0; i < LD_PER_THREAD; ++i) {
      const int idx = i * BLOCK + tid;
      float* dst = &tile[buf][(idx >> 4) * LDS_STRIDE + ((idx & 15) << 2)];
      dst[0] = tmp[i].x; dst[1] = tmp[i].y; dst[2] = tmp[i].z; dst[3] = tmp[i].w;
    }
#endif
  };

  issue_chunk(0, 0);

#pragma unroll 1
  for (int c = 0; c < NCHUNK; ++c) {
    const int buf = c & 1;
    if (c + 1 < NCHUNK) {
      issue_chunk((c + 1) * KC, buf ^ 1);       // buf^1 was released last iter
#if USE_ASYNC_LDS
      asm volatile("s_wait_asynccnt 0x8" ::: "memory");  // chunk c complete
#endif
    } else {
#if USE_ASYNC_LDS
      asm volatile("s_wait_asynccnt 0x0" ::: "memory");
#endif
    }
    __syncthreads();                            // all waves' chunk-c data ready

    const int k0 = c * KC;
    const float* arow = &tile[buf][(wave * 16 + m) * LDS_STRIDE];
#pragma unroll
    for (int kk = 0; kk < KC; kk += 4) {
      v2f a, b;
      a.x = arow[kk + koff];
      a.y = arow[kk + koff + 1];
      b.x = vsh[k0 + kk + koff];                // broadcast column
      b.y = vsh[k0 + kk + koff + 1];
      acc = __builtin_amdgcn_wmma_f32_16x16x4_f32(
          false, a, false, b, (short)0, acc, false, false);
    }
    __syncthreads();                            // release tile[buf]
  }

  // D layout: lane 0 -> (M=r, N=0) in acc[r]; lane 16 -> (M=r+8, N=0)
  if (lane == 0 || lane == 16) {
    float* o = &energies[rowBase + wave * 16 + ((lane == 16) ? 8 : 0)];
#pragma unroll
    for (int r = 0; r < 8; ++r) o[r] = acc[r];
  }
}

// ---------------------------------------------------------------------------
// Kernel C: softmax over 32768 energies, single block (traffic is negligible).
// ---------------------------------------------------------------------------
__global__ void __launch_bounds__(1024)
softmax_kernel(const float* __restrict__ e, float* __restrict__ out) {
  __shared__ float red[1024];
  const int tid = threadIdx.x;

  float mx = -3.402823466e38f;
  for (int i = tid; i < SEQ_LEN; i += 1024) mx = fmaxf(mx, e[i]);
  red[tid] = mx;
  __syncthreads();
  for (int s = 512; s > 0; s >>= 1) {
    if (tid < s) red[tid] = fmaxf(red[tid], red[tid + s]);
    __syncthreads();
  }
  const float gmax = red[0];
  __syncthreads();

  float sum = 0.f;
  for (int i = tid; i < SEQ_LEN; i += 1024) sum += expf(e[i] - gmax);
  red[tid] = sum;
  __syncthreads();
  for (int s = 512; s > 0; s >>= 1) {
    if (tid < s) red[tid] += red[tid + s];
    __syncthreads();
  }
  const float inv = 1.0f / red[0];

  for (int i = tid; i < SEQ_LEN; i += 1024) out[i] = expf(e[i] - gmax) * inv;
}

// ---------------------------------------------------------------------------
extern "C" void kernel_launch(void* const* d_in, const int* in_sizes, int n_in,
                              void* d_out, int out_size, void* d_ws, size_t ws_size,
                              hipStream_t stream) {
  (void)in_sizes; (void)n_in; (void)out_size; (void)ws_size;

  const float* hidden = (const float*)d_in[0];   // [1,1,1024]
  const float* enc    = (const float*)d_in[1];   // [32768,1024]
  const float* W      = (const float*)d_in[2];   // [1024,1024]
  // d_in[3] = b: constant shift of all energies -> cancels in softmax. Unused.

  float* v        = (float*)d_ws;                // [1024]
  float* energies = v + HID;                     // [32768]
  float* out      = (float*)d_out;               // [1,1,32768]

  proj_vec_kernel<<<HID / 256, 256, 0, stream>>>(W, hidden, v);
  energies_kernel<<<SEQ_LEN / ROWS_PER_WG, BLOCK, 0, stream>>>(enc, v, energies);
  softmax_kernel<<<1, 1024, 0, stream>>>(energies, out);
}